// GCN_1layer_89472758710373
// MI455X (gfx1250) — compile-verified
//
#include <hip/hip_runtime.h>
#include <hip/hip_bf16.h>

typedef __attribute__((ext_vector_type(2))) float v2f;
typedef __attribute__((ext_vector_type(8))) float v8f;

#define FEAT 128            // IN == OUT == 128
#define LDS_PITCH 132       // 128 + 4 pad to spread LDS banks
#define ROWS_PER_BLOCK 64   // 4 row-tiles of 16

// ---------------------------------------------------------------------------
// 1. degree init: every node has a self-loop -> deg starts at 1
// ---------------------------------------------------------------------------
__global__ void gcn_init_deg(float* __restrict__ deg, int n) {
    int i = blockIdx.x * blockDim.x + threadIdx.x;
    if (i < n) deg[i] = 1.0f;
}

// ---------------------------------------------------------------------------
// 2. degree accumulation over edge destinations
// ---------------------------------------------------------------------------
__global__ void gcn_accum_deg(const long long* __restrict__ dst,
                              float* __restrict__ deg, int e) {
    int i = blockIdx.x * blockDim.x + threadIdx.x;
    if (i < e) atomicAdd(&deg[(int)dst[i]], 1.0f);
}

// ---------------------------------------------------------------------------
// 3. dinv = deg^-1/2  (deg >= 1 always because of self-loops)
// ---------------------------------------------------------------------------
__global__ void gcn_dinv(const float* __restrict__ deg,
                         float* __restrict__ dinv, int n) {
    int i = blockIdx.x * blockDim.x + threadIdx.x;
    if (i < n) dinv[i] = rsqrtf(deg[i]);
}

// ---------------------------------------------------------------------------
// 4. x = node_feature @ W  via V_WMMA_F32_16X16X4_F32 (exact fp32 math)
//    256 threads = 8 waves; block handles 64 rows x 128 cols.
//    W (64KB) + A-tile (33KB) staged in dynamic LDS.
//    K-outer loop with 4 live accumulators: each B fragment is loaded from
//    LDS once and reused by 4 independent WMMAs (hides XDL latency, 4x less
//    B-side DS traffic).
// ---------------------------------------------------------------------------
__global__ void gcn_gemm_wmma(const float* __restrict__ X,   // [n,128]
                              const float* __restrict__ W,   // [128,128]
                              float* __restrict__ Y,         // [n,128]
                              int n) {
    extern __shared__ float lds[];
    float* Ws = lds;                               // 128 x LDS_PITCH
    float* As = lds + FEAT * LDS_PITCH;            // 64  x LDS_PITCH

    const int tid  = threadIdx.x;
    const int wave = tid >> 5;          // 0..7 -> column tile
    const int lane = tid & 31;
    const int row0 = blockIdx.x * ROWS_PER_BLOCK;

    // cooperative load of W: 128x128 floats = 4096 float4, 16 per thread
    for (int i = tid; i < FEAT * (FEAT / 4); i += 256) {
        int r  = i >> 5;                // row
        int c4 = i & 31;                // float4 within row
        float4 v = ((const float4*)(W + r * FEAT))[c4];
        float* d = Ws + r * LDS_PITCH + c4 * 4;
        d[0] = v.x; d[1] = v.y; d[2] = v.z; d[3] = v.w;
    }
    // cooperative load of the 64-row A tile (zero-pad past n)
    for (int i = tid; i < ROWS_PER_BLOCK * (FEAT / 4); i += 256) {
        int r    = i >> 5;
        int c4   = i & 31;
        int grow = row0 + r;
        float4 v = make_float4(0.f, 0.f, 0.f, 0.f);
        if (grow < n) v = ((const float4*)(X + (size_t)grow * FEAT))[c4];
        float* d = As + r * LDS_PITCH + c4 * 4;
        d[0] = v.x; d[1] = v.y; d[2] = v.z; d[3] = v.w;
    }
    __syncthreads();

    // A 16x4 f32 frag: lane m = lane&15 holds row m; half-wave selects K pair
    const int m     = lane & 15;
    const int khalf = (lane >> 4) << 1;   // 0 or 2
    const int col0  = wave * 16;          // this wave's 16 output columns

    v8f acc0 = {}, acc1 = {}, acc2 = {}, acc3 = {};
    const float* arow = As + m * LDS_PITCH;          // row-tile rt at +rt*16*PITCH

    for (int k0 = 0; k0 < FEAT; k0 += 4) {
        // B 4x16 f32 frag: lane n = lane&15 holds col n; half-wave K pair
        v2f b;
        b.x = Ws[(k0 + khalf)     * LDS_PITCH + col0 + m];
        b.y = Ws[(k0 + khalf + 1) * LDS_PITCH + col0 + m];

        v2f a0, a1, a2, a3;
        a0.x = arow[k0 + khalf];
        a0.y = arow[k0 + khalf + 1];
        a1.x = arow[16 * LDS_PITCH + k0 + khalf];
        a1.y = arow[16 * LDS_PITCH + k0 + khalf + 1];
        a2.x = arow[32 * LDS_PITCH + k0 + khalf];
        a2.y = arow[32 * LDS_PITCH + k0 + khalf + 1];
        a3.x = arow[48 * LDS_PITCH + k0 + khalf];
        a3.y = arow[48 * LDS_PITCH + k0 + khalf + 1];

        acc0 = __builtin_amdgcn_wmma_f32_16x16x4_f32(
            false, a0, false, b, (short)0, acc0, false, false);
        acc1 = __builtin_amdgcn_wmma_f32_16x16x4_f32(
            false, a1, false, b, (short)0, acc1, false, false);
        acc2 = __builtin_amdgcn_wmma_f32_16x16x4_f32(
            false, a2, false, b, (short)0, acc2, false, false);
        acc3 = __builtin_amdgcn_wmma_f32_16x16x4_f32(
            false, a3, false, b, (short)0, acc3, false, false);
    }

    // C/D layout: VGPR j -> M = j + 8*(lane>=16), N = lane&15
    const int nn    = lane & 15;
    const int mbase = (lane >> 4) * 8;
    v8f accs[4] = {acc0, acc1, acc2, acc3};
#pragma unroll
    for (int rt = 0; rt < 4; ++rt) {
        const int grow0 = row0 + rt * 16;
#pragma unroll
        for (int j = 0; j < 8; ++j) {
            int gr = grow0 + mbase + j;
            if (gr < n) Y[(size_t)gr * FEAT + col0 + nn] = accs[rt][j];
        }
    }
}

// ---------------------------------------------------------------------------
// 5. self-loop term: out[i,:] = x[i,:] * dinv[i]^2  (fully overwrites d_out)
// ---------------------------------------------------------------------------
__global__ void gcn_selfloop_init(const float* __restrict__ x,
                                  const float* __restrict__ dinv,
                                  float* __restrict__ out, int n) {
    int i = blockIdx.x * blockDim.x + threadIdx.x;   // over n * 32 float4s
    int total = n * (FEAT / 4);
    if (i < total) {
        int row = i >> 5;
        float di = dinv[row];
        float s = di * di;
        float4 v = ((const float4*)x)[i];
        float4 r = make_float4(v.x * s, v.y * s, v.z * s, v.w * s);
        ((float4*)out)[i] = r;
    }
}

// ---------------------------------------------------------------------------
// 6. edge scatter: one wave32 per edge; lane L handles 4 consecutive floats.
//    Gathers are 512B coalesced per wave; x lives in L2 (51MB < 192MB).
// ---------------------------------------------------------------------------
__global__ void gcn_scatter(const float* __restrict__ x,
                            const float* __restrict__ dinv,
                            const long long* __restrict__ src,
                            const long long* __restrict__ dst,
                            float* __restrict__ out, int e) {
    int gw   = (blockIdx.x * blockDim.x + threadIdx.x) >> 5;  // edge id
    int lane = threadIdx.x & 31;
    if (gw >= e) return;
    int s = (int)src[gw];
    int d = (int)dst[gw];
    float norm = dinv[s] * dinv[d];
    float4 v = ((const float4*)(x + (size_t)s * FEAT))[lane];
    float* o = out + (size_t)d * FEAT + lane * 4;
    atomicAdd(o + 0, v.x * norm);
    atomicAdd(o + 1, v.y * norm);
    atomicAdd(o + 2, v.z * norm);
    atomicAdd(o + 3, v.w * norm);
}

// ---------------------------------------------------------------------------
// 7. out = relu(out + b)
// ---------------------------------------------------------------------------
__global__ void gcn_bias_relu(float* __restrict__ out,
                              const float* __restrict__ b, int n) {
    int i = blockIdx.x * blockDim.x + threadIdx.x;   // over n * 32 float4s
    int total = n * (FEAT / 4);
    if (i < total) {
        int c4 = i & 31;
        float4 bb = ((const float4*)b)[c4];
        float4 v = ((float4*)out)[i];
        v.x = fmaxf(v.x + bb.x, 0.f);
        v.y = fmaxf(v.y + bb.y, 0.f);
        v.z = fmaxf(v.z + bb.z, 0.f);
        v.w = fmaxf(v.w + bb.w, 0.f);
        ((float4*)out)[i] = v;
    }
}

// ---------------------------------------------------------------------------
extern "C" void kernel_launch(void* const* d_in, const int* in_sizes, int n_in,
                              void* d_out, int out_size, void* d_ws, size_t ws_size,
                              hipStream_t stream) {
    const float*     Xin  = (const float*)d_in[0];       // [n,128]
    const long long* eidx = (const long long*)d_in[1];   // [2,e] int64
    const float*     W    = (const float*)d_in[2];       // [128,128]
    const float*     b    = (const float*)d_in[3];       // [128]

    const int n = in_sizes[0] / FEAT;
    const int e = in_sizes[1] / 2;
    const long long* src = eidx;
    const long long* dst = eidx + e;

    // workspace layout: x [n*128] | deg [n] | dinv [n]
    float* xw   = (float*)d_ws;
    float* deg  = xw + (size_t)n * FEAT;
    float* dinv = deg + n;

    float* out = (float*)d_out;

    // 1-3: normalization factors
    gcn_init_deg<<<(n + 255) / 256, 256, 0, stream>>>(deg, n);
    gcn_accum_deg<<<(e + 255) / 256, 256, 0, stream>>>(dst, deg, e);
    gcn_dinv<<<(n + 255) / 256, 256, 0, stream>>>(deg, dinv, n);

    // 4: WMMA GEMM
    {
        size_t ldsBytes = (size_t)(FEAT + ROWS_PER_BLOCK) * LDS_PITCH * sizeof(float);
        int grid = (n + ROWS_PER_BLOCK - 1) / ROWS_PER_BLOCK;
        gcn_gemm_wmma<<<grid, 256, ldsBytes, stream>>>(Xin, W, xw, n);
    }

    // 5: self-loop term initializes d_out
    {
        int total = n * (FEAT / 4);
        gcn_selfloop_init<<<(total + 255) / 256, 256, 0, stream>>>(xw, dinv, out, n);
    }

    // 6: edge scatter (1 wave per edge -> 32 threads per edge)
    {
        long long threads = (long long)e * 32;
        int grid = (int)((threads + 255) / 256);
        gcn_scatter<<<grid, 256, 0, stream>>>(xw, dinv, src, dst, out, e);
    }

    // 7: bias + relu
    {
        int total = n * (FEAT / 4);
        gcn_bias_relu<<<(total + 255) / 256, 256, 0, stream>>>(out, b, n);
    }
}